// LUT_Symmetric_80032420594226
// MI455X (gfx1250) — compile-verified
//
#include <hip/hip_runtime.h>
#include <hip/hip_bf16.h>

// LUT-symmetric 4x4 tile expansion. Pure gather/permute workload:
// store-bandwidth bound (~137 MB HBM -> ~6us at 23.3 TB/s). No matmul =>
// WMMA not applicable (rot90 is a per-lane permutation; WMMA is wave-wide).
// CDNA5 features used: NT temporal-hint stores/loads (preserve L2 residency
// of the 1MB weight tables against the 128MB output stream), LDS-served
// symmetry table (ds_load_u8).

#define BLK 256
#define LBASE 17u

typedef __attribute__((ext_vector_type(4))) float f4;

// SYM[r0*64 + r1*16 + r2*4 + r3] = ptype | (rot<<2) | (flip<<4)
// Only the 24 permutation keys are ever read; filler = 0.
__constant__ unsigned char SYM[256] = {
    //  0- 15
    0,0,0,0,0,0,0,0,0,0,0,0,0,0,0,0,
    // 16- 31 : 27->0(t0 r0 f0), 30->1(t1 r0 f0)
    0,0,0,0,0,0,0,0,0,0,0,0,0,0,1,0,
    // 32- 47 : 39->20(t0 r1 f1), 45->2(t2 r0 f0)
    0,0,0,0,0,0,0,20,0,0,0,0,0,2,0,0,
    // 48- 63 : 54->21(t1 r1 f1), 57->22(t2 r1 f1)
    0,0,0,0,0,0,21,0,0,22,0,0,0,0,0,0,
    // 64- 79 : 75->17(t1 r0 f1), 78->16(t0 r0 f1)
    0,0,0,0,0,0,0,0,0,0,0,17,0,0,16,0,
    // 80- 95
    0,0,0,0,0,0,0,0,0,0,0,0,0,0,0,0,
    // 96-111 : 99->5(t1 r1 f0), 108->30(t2 r3 f1)
    0,0,0,5,0,0,0,0,0,0,0,0,30,0,0,0,
    //112-127 : 114->4(t0 r1 f0), 120->10(t2 r2 f0)
    0,0,4,0,0,0,0,0,10,0,0,0,0,0,0,0,
    //128-143 : 135->18(t2 r0 f1), 141->12(t0 r3 f0)
    0,0,0,0,0,0,0,18,0,0,0,0,0,12,0,0,
    //144-159 : 147->6(t2 r1 f0), 156->29(t1 r3 f1)
    0,0,0,6,0,0,0,0,0,0,0,0,29,0,0,0,
    //160-175
    0,0,0,0,0,0,0,0,0,0,0,0,0,0,0,0,
    //176-191 : 177->24(t0 r2 f1), 180->9(t1 r2 f0)
    0,24,0,0,9,0,0,0,0,0,0,0,0,0,0,0,
    //192-207 : 198->14(t2 r3 f0), 201->13(t1 r3 f0)
    0,0,0,0,0,0,14,0,0,13,0,0,0,0,0,0,
    //208-223 : 210->26(t2 r2 f1), 216->28(t0 r3 f1)
    0,0,26,0,0,0,0,0,28,0,0,0,0,0,0,0,
    //224-239 : 225->25(t1 r2 f1), 228->8(t0 r2 f0)
    0,25,0,0,8,0,0,0,0,0,0,0,0,0,0,0,
    //240-255
    0,0,0,0,0,0,0,0,0,0,0,0,0,0,0,0
};

__device__ __forceinline__ float round_clip(float x) {
    x = __builtin_rintf(x);              // v_rndne_f32 (round half to even, matches jnp.round)
    x = fmaxf(x, -128.0f);
    return fminf(x, 127.0f);
}

__global__ __launch_bounds__(BLK) void lut_sym_kernel(
    const int*   __restrict__ idxv,
    const float* __restrict__ w0,
    const float* __restrict__ w1,
    const float* __restrict__ w2,
    float*       __restrict__ out,
    int n)
{
    __shared__ unsigned char sym_lds[256];
    sym_lds[threadIdx.x] = SYM[threadIdx.x];   // BLK == 256
    __syncthreads();

    int gid = blockIdx.x * BLK + threadIdx.x;
    if (gid >= n) return;

    // ---- base-17 digit decomposition (d fastest) ----
    unsigned int i  = (unsigned int)__builtin_nontemporal_load(idxv + gid);
    unsigned int dd = i % LBASE; i /= LBASE;
    unsigned int cc = i % LBASE; i /= LBASE;
    unsigned int bb = i % LBASE;
    unsigned int aa = i / LBASE;

    // ---- stable sort of (a,b,c,d) via keys value*4+pos, 5-comparator network ----
    unsigned int k0 = aa * 4u + 0u;
    unsigned int k1 = bb * 4u + 1u;
    unsigned int k2 = cc * 4u + 2u;
    unsigned int k3 = dd * 4u + 3u;
    unsigned int t;
    if (k0 > k1) { t = k0; k0 = k1; k1 = t; }
    if (k2 > k3) { t = k2; k2 = k3; k3 = t; }
    if (k0 > k2) { t = k0; k0 = k2; k2 = t; }
    if (k1 > k3) { t = k1; k1 = k3; k3 = t; }
    if (k1 > k2) { t = k1; k1 = k2; k2 = t; }

    unsigned int s0 = k0 >> 2, s1 = k1 >> 2, s2 = k2 >> 2, s3 = k3 >> 2; // sorted values
    // ranks[order[j]] = j ; symkey = r0*64 + r1*16 + r2*4 + r3
    unsigned int symkey = 1u * (64u >> (2u * (k1 & 3u)))
                        + 2u * (64u >> (2u * (k2 & 3u)))
                        + 3u * (64u >> (2u * (k3 & 3u)));

    unsigned int packed = sym_lds[symkey];     // ds_load_u8
    unsigned int ptype  = packed & 3u;
    unsigned int rot    = (packed >> 2) & 3u;
    unsigned int flip   = packed >> 4;

    // ---- closed-form IDX_MAP: lexicographic rank of non-decreasing 4-tuple ----
    // widx = [C(20,4)-C(20-a,4)] + [C(19-a,3)-C(19-b,3)] + [C(18-b,2)-C(18-c,2)] + (d-c)
    int A = 20 - (int)s0;
    int B = 19 - (int)s1;
    int C = 18 - (int)s2;
    int widx = 4845
             - (A * (A - 1) * (A - 2) * (A - 3)) / 24
             + ((A - 1) * (A - 2) * (A - 3)) / 6
             - (B * (B - 1) * (B - 2)) / 6
             + ((B - 1) * (B - 2)) / 2
             - (C * (C - 1)) / 2
             + (int)s3 - (int)s2;

    // ---- gather 4x4 tile (regular loads: keep weight tables hot in L2) ----
    const float* wbase = (ptype == 0u) ? w0 : ((ptype == 1u) ? w1 : w2);
    const f4* wrow = reinterpret_cast<const f4*>(wbase + (size_t)widx * 16u);
    f4 q0 = wrow[0], q1 = wrow[1], q2 = wrow[2], q3 = wrow[3];

    float r[4][4];
    r[0][0] = round_clip(q0.x); r[0][1] = round_clip(q0.y); r[0][2] = round_clip(q0.z); r[0][3] = round_clip(q0.w);
    r[1][0] = round_clip(q1.x); r[1][1] = round_clip(q1.y); r[1][2] = round_clip(q1.z); r[1][3] = round_clip(q1.w);
    r[2][0] = round_clip(q2.x); r[2][1] = round_clip(q2.y); r[2][2] = round_clip(q2.z); r[2][3] = round_clip(q2.w);
    r[3][0] = round_clip(q3.x); r[3][1] = round_clip(q3.y); r[3][2] = round_clip(q3.z); r[3][3] = round_clip(q3.w);

    // ---- flip (reverse x) branch-free ----
    float v[4][4];
#pragma unroll
    for (int y = 0; y < 4; ++y) {
        v[y][0] = flip ? r[y][3] : r[y][0];
        v[y][1] = flip ? r[y][2] : r[y][1];
        v[y][2] = flip ? r[y][1] : r[y][2];
        v[y][3] = flip ? r[y][0] : r[y][3];
    }

    // ---- rot90 CCW by `rot` in (row,col): out[i][j] = v[...] ----
    f4 e0, e1, e2, e3;
    if (rot == 0u) {
        e0 = (f4){v[0][0], v[0][1], v[0][2], v[0][3]};
        e1 = (f4){v[1][0], v[1][1], v[1][2], v[1][3]};
        e2 = (f4){v[2][0], v[2][1], v[2][2], v[2][3]};
        e3 = (f4){v[3][0], v[3][1], v[3][2], v[3][3]};
    } else if (rot == 1u) {            // out[i][j] = v[j][3-i]
        e0 = (f4){v[0][3], v[1][3], v[2][3], v[3][3]};
        e1 = (f4){v[0][2], v[1][2], v[2][2], v[3][2]};
        e2 = (f4){v[0][1], v[1][1], v[2][1], v[3][1]};
        e3 = (f4){v[0][0], v[1][0], v[2][0], v[3][0]};
    } else if (rot == 2u) {            // out[i][j] = v[3-i][3-j]
        e0 = (f4){v[3][3], v[3][2], v[3][1], v[3][0]};
        e1 = (f4){v[2][3], v[2][2], v[2][1], v[2][0]};
        e2 = (f4){v[1][3], v[1][2], v[1][1], v[1][0]};
        e3 = (f4){v[0][3], v[0][2], v[0][1], v[0][0]};
    } else {                           // out[i][j] = v[3-j][i]
        e0 = (f4){v[3][0], v[2][0], v[1][0], v[0][0]};
        e1 = (f4){v[3][1], v[2][1], v[1][1], v[0][1]};
        e2 = (f4){v[3][2], v[2][2], v[1][2], v[0][2]};
        e3 = (f4){v[3][3], v[2][3], v[1][3], v[0][3]};
    }

    // ---- streaming NT stores: don't pollute L2 (keep weight tables resident) ----
    f4* dst = reinterpret_cast<f4*>(out + (size_t)gid * 16u);
    __builtin_nontemporal_store(e0, dst + 0);
    __builtin_nontemporal_store(e1, dst + 1);
    __builtin_nontemporal_store(e2, dst + 2);
    __builtin_nontemporal_store(e3, dst + 3);
}

extern "C" void kernel_launch(void* const* d_in, const int* in_sizes, int n_in,
                              void* d_out, int out_size, void* d_ws, size_t ws_size,
                              hipStream_t stream) {
    const int*   index = (const int*)  d_in[0];
    const float* w0    = (const float*)d_in[1];
    const float* w1    = (const float*)d_in[2];
    const float* w2    = (const float*)d_in[3];
    float*       out   = (float*)d_out;
    int n = in_sizes[0];
    int blocks = (n + BLK - 1) / BLK;
    hipLaunchKernelGGL(lut_sym_kernel, dim3(blocks), dim3(BLK), 0, stream,
                       index, w0, w1, w2, out, n);
}